// DigitSelector_15032385536393
// MI455X (gfx1250) — compile-verified
//
#include <hip/hip_runtime.h>
#include <cstdint>

typedef float v2f  __attribute__((ext_vector_type(2)));
typedef float v8f  __attribute__((ext_vector_type(8)));
typedef unsigned int u32x4 __attribute__((ext_vector_type(4)));
typedef int  i32x8 __attribute__((ext_vector_type(8)));
typedef int  i32x4 __attribute__((ext_vector_type(4)));

// ---- LDS layout (float offsets) ----
// W_q staged by TDM with padding: 64 rows, logical 512 cols, padded stride 516
// (pad_interval=256 DW, pad_amount=2 DW -> element (r,c) at r*516 + c + 2*(c>=256))
#define WQ_OFF      0
#define WQ_STRIDE   516              // 64*516 = 33024 floats
#define SLOT_OFF    33024            // 16 x 64, padded stride 68 -> 1088 floats
#define SLOT_STRIDE 68
#define WK_OFF      34112            // 64
#define WV_OFF      34176            // 64
#define Q_OFF       34240            // per-wave 16x64 padded stride 68 -> 1088 * 8 waves
#define Q_STRIDE    68
#define WATT_OFF    42944            // per-wave 16x16 attn weights, stride 17 -> 272 * 8
#define WD_OFF      45120            // per-wave 16 floats -> 16 * 8
#define SMEM_FLOATS 45248            // 180,992 bytes < 320 KB

__global__ __launch_bounds__(256)
void digit_selector_kernel(const float* __restrict__ digits,
                           const float* __restrict__ sgn,
                           const float* __restrict__ pos_emb,
                           const int*   __restrict__ offset,
                           const float* __restrict__ W_q,
                           const float* __restrict__ w_k,
                           const float* __restrict__ w_v,
                           const float* __restrict__ slot_embed,
                           float* __restrict__ out)
{
    extern __shared__ float smem[];
    const int tid  = threadIdx.x;
    const int wid  = tid >> 5;
    const int lane = tid & 31;
    const int half = lane >> 4;     // 0|1
    const int nsl  = lane & 15;     // slot / column / row-in-tile index

    // ---------------- setup: TDM W_q -> LDS, slot/wk/wv via VMEM ----------------
    if (wid == 0) {
        // Tensor DMA descriptor: 2D tensor 512x64 fp32, tile = whole tensor,
        // LDS padding: 2 DWORDs every 256 DWORDs -> padded row stride 516 DW.
        unsigned lds_addr = (unsigned)(size_t)&smem[WQ_OFF];
        unsigned long long ga = (unsigned long long)(size_t)W_q;
        u32x4 g0;
        g0[0] = 1u;                                              // count=1 (valid)
        g0[1] = lds_addr;                                        // lds_addr
        g0[2] = (unsigned)(ga & 0xFFFFFFFFull);                  // global_addr lo
        g0[3] = (unsigned)((ga >> 32) & 0x1FFFFFFull) | (2u << 30); // ga hi | type=2
        i32x8 g1;
        g1[0] = (2 << 16)        // data_size = 4 bytes
              | (1 << 20)        // pad_enable
              | (7 << 22)        // pad_interval code 7 = 256 DWORDs
              | (1 << 25);       // pad_amount  code 1 = 2 DWORDs
        g1[1] = (int)(512u << 16);   // [63:48]  tensor_dim0 lo16 = 512
        g1[2] = (int)(64u  << 16);   // [79:64] dim0 hi=0, [95:80] tensor_dim1 lo16 = 64
        g1[3] = (int)(512u << 16);   // [111:96] dim1 hi=0, [127:112] tile_dim0 = 512
        g1[4] = 64;                  // [143:128] tile_dim1 = 64, tile_dim2 = 0
        g1[5] = 512;                 // tensor_dim0_stride lo32 = 512
        g1[6] = 0;
        g1[7] = 0;
        i32x4 gz4 = {0, 0, 0, 0};    // 2D: groups 2/3 unused
        i32x8 gz8 = {0, 0, 0, 0, 0, 0, 0, 0};
        __builtin_amdgcn_tensor_load_to_lds(g0, g1, gz4, gz4, gz8, 0);
    }
    {
        // slot_embed: 16x64 -> padded stride 68 (bank-conflict-free B reads)
        int idx = tid * 4;
        int r = idx >> 6, c = idx & 63;
        float4 v = *(const float4*)(slot_embed + idx);
        *(float4*)&smem[SLOT_OFF + r * SLOT_STRIDE + c] = v;
        if (tid < 64)       smem[WK_OFF + tid]        = w_k[tid];
        else if (tid < 128) smem[WV_OFF + (tid - 64)] = w_v[tid - 64];
    }
    if (wid == 0) __builtin_amdgcn_s_wait_tensorcnt(0);
    __syncthreads();

    const long long r0 = ((long long)blockIdx.x * 8 + wid) * 16;   // wave's row tile
    float* qlds  = &smem[Q_OFF    + wid * (16 * Q_STRIDE)];
    float* wlds  = &smem[WATT_OFF + wid * (16 * 17)];
    float* wdlds = &smem[WD_OFF   + wid * 16];

    // ------------- phase A: q = pos_emb x W_q^T  (v_wmma_f32_16x16x4_f32) -------------
    // A frag: lane L holds row (L&15), K = k0 + 2*(L>>4) + {0,1}  (16x4 f32 layout)
    // fused: each loaded float2 of pos_emb is also the output passthrough copy.
    v8f q0 = {}, q1 = {}, q2 = {}, q3 = {};
    const float* Ap = pos_emb + (r0 + nsl) * 512 + 2 * half;
    float*       Cp = out     + (r0 + nsl) * 578 + 66 + 2 * half;   // 8B aligned
    #pragma unroll 4
    for (int k0 = 0; k0 < 512; k0 += 4) {
        v2f a = *(const v2f*)(Ap + k0);
        *(v2f*)(Cp + k0) = a;                                   // free pos_emb copy
        const int kadj = k0 + 2 * half + ((k0 >= 256) ? 2 : 0); // TDM mid-row pad
        v2f b0 = *(const v2f*)&smem[WQ_OFF + ( 0 + nsl) * WQ_STRIDE + kadj];
        v2f b1 = *(const v2f*)&smem[WQ_OFF + (16 + nsl) * WQ_STRIDE + kadj];
        v2f b2 = *(const v2f*)&smem[WQ_OFF + (32 + nsl) * WQ_STRIDE + kadj];
        v2f b3 = *(const v2f*)&smem[WQ_OFF + (48 + nsl) * WQ_STRIDE + kadj];
        q0 = __builtin_amdgcn_wmma_f32_16x16x4_f32(false, a, false, b0, (short)0, q0, false, false);
        q1 = __builtin_amdgcn_wmma_f32_16x16x4_f32(false, a, false, b1, (short)0, q1, false, false);
        q2 = __builtin_amdgcn_wmma_f32_16x16x4_f32(false, a, false, b2, (short)0, q2, false, false);
        q3 = __builtin_amdgcn_wmma_f32_16x16x4_f32(false, a, false, b3, (short)0, q3, false, false);
    }

    // ------------- phase B: q tile (D layout) -> LDS (row-major, padded) -------------
    #pragma unroll
    for (int j = 0; j < 8; j++) {
        int m = j + 8 * half;
        qlds[m * Q_STRIDE +  0 + nsl] = q0[j];
        qlds[m * Q_STRIDE + 16 + nsl] = q1[j];
        qlds[m * Q_STRIDE + 32 + nsl] = q2[j];
        qlds[m * Q_STRIDE + 48 + nsl] = q3[j];
    }
    __syncthreads();

    // ------------- phase C: qk[row] = q . w_k (half-wave split + shfl) -------------
    float part = 0.f;
    {
        const float* qrow = &qlds[nsl * Q_STRIDE + 32 * half];
        const float* wkp  = &smem[WK_OFF + 32 * half];
        #pragma unroll
        for (int a = 0; a < 32; a++) part = fmaf(qrow[a], wkp[a], part);
    }
    const float qk = part + __shfl_xor(part, 16, 32);   // lane (L&15) holds qk[row L&15]

    // ------------- phase D: slot scores = q[16x64] x slot_embed^T[64x16] -------------
    v8f sc = {};
    #pragma unroll
    for (int a0 = 0; a0 < 64; a0 += 4) {
        v2f qa = *(const v2f*)&qlds[nsl * Q_STRIDE + a0 + 2 * half];
        v2f sb = *(const v2f*)&smem[SLOT_OFF + nsl * SLOT_STRIDE + a0 + 2 * half];
        sc = __builtin_amdgcn_wmma_f32_16x16x4_f32(false, qa, false, sb, (short)0, sc, false, false);
    }

    // ------------- phase E: mask + softmax over K=16 (cross-lane in 16-lane halves) ---
    const float inv_scale = 0.125f;   // 1/sqrt(64)
    #pragma unroll
    for (int j = 0; j < 8; j++) {
        const int m = j + 8 * half;                       // row in tile
        float d = digits[(r0 + m) * 16 + nsl];
        float qkm = __shfl(qk, m, 32);                    // broadcast qk[m]
        float s = fmaf(d, qkm, sc[j]) * inv_scale;
        if (d < 0.f) s = -1e9f;
        float mx = s;
        mx = fmaxf(mx, __shfl_xor(mx, 1, 32));
        mx = fmaxf(mx, __shfl_xor(mx, 2, 32));
        mx = fmaxf(mx, __shfl_xor(mx, 4, 32));
        mx = fmaxf(mx, __shfl_xor(mx, 8, 32));
        float e  = __expf(s - mx);
        float se = e, sd = e * d;
        se += __shfl_xor(se, 1, 32);  sd += __shfl_xor(sd, 1, 32);
        se += __shfl_xor(se, 2, 32);  sd += __shfl_xor(sd, 2, 32);
        se += __shfl_xor(se, 4, 32);  sd += __shfl_xor(sd, 4, 32);
        se += __shfl_xor(se, 8, 32);  sd += __shfl_xor(sd, 8, 32);
        wlds[m * 17 + nsl] = e / se;                      // attn weight
        if (nsl == 0) wdlds[m] = sd / se;                 // sum_k w_k * d_k
    }
    __syncthreads();

    // ------------- phase F: ctx = wd*w_v + attn_w @ slot_embed, + scalars -------------
    {
        const int row = lane >> 1;              // 2 lanes per row
        const int c0  = (lane & 1) * 32;        // 32 cols per lane
        float wr[16];
        #pragma unroll
        for (int n = 0; n < 16; n++) wr[n] = wlds[row * 17 + n];
        const float wd = wdlds[row];
        float4 acc[8];
        #pragma unroll
        for (int t = 0; t < 8; t++) {
            float4 wv4 = *(const float4*)&smem[WV_OFF + c0 + 4 * t];
            acc[t].x = wd * wv4.x; acc[t].y = wd * wv4.y;
            acc[t].z = wd * wv4.z; acc[t].w = wd * wv4.w;
        }
        #pragma unroll
        for (int n = 0; n < 16; n++) {
            const float wn = wr[n];
            #pragma unroll
            for (int t = 0; t < 8; t++) {
                float4 sv = *(const float4*)&smem[SLOT_OFF + n * SLOT_STRIDE + c0 + 4 * t];
                acc[t].x = fmaf(wn, sv.x, acc[t].x);
                acc[t].y = fmaf(wn, sv.y, acc[t].y);
                acc[t].z = fmaf(wn, sv.z, acc[t].z);
                acc[t].w = fmaf(wn, sv.w, acc[t].w);
            }
        }
        float* op = out + (r0 + row) * 578 + 1 + c0;   // ctx at cols [1,65)
        #pragma unroll
        for (int t = 0; t < 8; t++) {
            op[4*t+0] = acc[t].x; op[4*t+1] = acc[t].y;
            op[4*t+2] = acc[t].z; op[4*t+3] = acc[t].w;
        }
    }
    if (lane < 16) {                           // d_hard (col 0) and sign (col 65)
        const long long gs = r0 + lane;
        int off = offset[gs];
        off = off < 0 ? 0 : (off > 15 ? 15 : off);
        float dh = digits[gs * 16 + off];
        out[gs * 578]      = dh < 0.f ? 0.f : dh;
        out[gs * 578 + 65] = sgn[gs];
    }
}

extern "C" void kernel_launch(void* const* d_in, const int* in_sizes, int n_in,
                              void* d_out, int out_size, void* d_ws, size_t ws_size,
                              hipStream_t stream) {
    (void)in_sizes; (void)n_in; (void)out_size; (void)d_ws; (void)ws_size;
    const float* digits = (const float*)d_in[0];
    const float* sgn    = (const float*)d_in[1];
    const float* pos    = (const float*)d_in[2];
    const int*   offs   = (const int*)  d_in[3];
    const float* W_q    = (const float*)d_in[4];
    const float* w_k    = (const float*)d_in[5];
    const float* w_v    = (const float*)d_in[6];
    const float* slot   = (const float*)d_in[7];
    float* out = (float*)d_out;

    // 131072 rows total, 128 rows per 256-thread block (8 waves x 16-row tiles)
    dim3 grid(1024), block(256);
    size_t smem_bytes = (size_t)SMEM_FLOATS * sizeof(float);
    digit_selector_kernel<<<grid, block, smem_bytes, stream>>>(
        digits, sgn, pos, offs, W_q, w_k, w_v, slot, out);
}